// DGCNN_1288490189417
// MI455X (gfx1250) — compile-verified
//
#include <hip/hip_runtime.h>
#include <hip/hip_bf16.h>
#include <math.h>

// ---- problem constants (match reference) ----
#define Nn   65536      // total nodes (G*NPG)
#define Ee   1048576    // edges
#define Gg   128        // graphs
#define NPGc 512        // nodes per graph
#define Hc   32         // hidden channels
#define FTc  97         // total latent dim (32*3 + 1)
#define Kc   30         // sort-pool k
#define C1c  16
#define C2c  32
#define KW2c 5
#define P1c  30         // conv1 out positions (== Kc)
#define PPc  15         // after maxpool
#define P2c  11         // conv2 out positions
#define DENSEc 352      // C2c * P2c

typedef __attribute__((ext_vector_type(2))) float v2f;
typedef __attribute__((ext_vector_type(8))) float v8f;

// ---------------- degree / normalization ----------------
__global__ void k_deg_init(float* deg) {
    int i = blockIdx.x * blockDim.x + threadIdx.x;
    if (i < Nn) deg[i] = 1.0f;   // self-loop weight
}
__global__ void k_deg_accum(const int* __restrict__ dst, float* deg) {
    int e = blockIdx.x * blockDim.x + threadIdx.x;
    if (e < Ee) atomicAdd(&deg[dst[e]], 1.0f);
}
__global__ void k_deg_rsqrt(float* deg) {
    int i = blockIdx.x * blockDim.x + threadIdx.x;
    if (i < Nn) deg[i] = rsqrtf(deg[i]);
}

// ---------------- embedding lookup ----------------
__global__ void k_embed(const int* __restrict__ z, const float* __restrict__ z_emb,
                        float* __restrict__ x0) {
    int t = blockIdx.x * blockDim.x + threadIdx.x;   // over N*32
    if (t >= Nn * Hc) return;
    int i = t >> 5, c = t & 31;
    x0[t] = z_emb[z[i] * Hc + c];
}

// ---------------- GCN dense part: H = X(Nx32) @ W(32x32) in exact f32 WMMA ----------------
// One wave per 16-row tile; K=32 via 8 steps of v_wmma_f32_16x16x4_f32, 2 col-tiles.
// Layouts per ISA 7.12.2:
//   A 16x4 (MxK):  lanes 0-15 (M=lane):  VGPR0=K0, VGPR1=K1; lanes 16-31: K2,K3
//   B 4x16 (KxN):  lane holds column N=lane&15; VGPR j = K row (lane-half selects K pair)
//   C/D 16x16 f32: VGPR r -> row r + 8*(lane>=16), col = lane&15
__global__ void k_gcn_matmul_wmma(const float* __restrict__ X, int ldx,
                                  const float* __restrict__ W,
                                  float* __restrict__ Hout) {
    const int lane = threadIdx.x & 31;
    const int warp = threadIdx.x >> 5;               // 4 waves per block
    const int rowBase = (blockIdx.x * 4 + warp) * 16;
    const int m  = lane & 15;
    const int hi = lane >> 4;                        // lane-half select

    const float* xr = X + (size_t)(rowBase + m) * ldx + 2 * hi;

    v8f acc0 = {}, acc1 = {};
#pragma unroll
    for (int s = 0; s < 8; ++s) {
        v2f a;
        a[0] = xr[4 * s];
        a[1] = xr[4 * s + 1];
        const int k = 4 * s + 2 * hi;
        v2f b0, b1;
        b0[0] = W[k * Hc + m];            b0[1] = W[(k + 1) * Hc + m];
        b1[0] = W[k * Hc + 16 + m];       b1[1] = W[(k + 1) * Hc + 16 + m];
        acc0 = __builtin_amdgcn_wmma_f32_16x16x4_f32(false, a, false, b0,
                                                     (short)0, acc0, false, false);
        acc1 = __builtin_amdgcn_wmma_f32_16x16x4_f32(false, a, false, b1,
                                                     (short)0, acc1, false, false);
    }
#pragma unroll
    for (int r = 0; r < 8; ++r) {
        int mr = r + hi * 8;
        Hout[(size_t)(rowBase + mr) * Hc + m]      = acc0[r];
        Hout[(size_t)(rowBase + mr) * Hc + 16 + m] = acc1[r];
    }
}

// ---------------- layer-3 dense part: h1 = X(Nx32) @ W3(32x1) ----------------
__global__ void k_matvec32(const float* __restrict__ X, int ldx,
                           const float* __restrict__ w, float* __restrict__ out) {
    int i = blockIdx.x * blockDim.x + threadIdx.x;
    if (i >= Nn) return;
    const float* xr = X + (size_t)i * ldx;
    float s = 0.f;
#pragma unroll
    for (int k = 0; k < Hc; ++k) s += xr[k] * w[k];
    out[i] = s;
}

// ---------------- edge scatter (32-ch): agg[dst] += dis[src]*dis[dst]*h[src] ----------------
// One thread per (edge, 4-channel group): one b128 load + 4 f32 atomics.
__global__ void k_edge_scatter32(const float* __restrict__ h, const float* __restrict__ dis,
                                 const int* __restrict__ src, const int* __restrict__ dst,
                                 float* __restrict__ agg) {
    int t = blockIdx.x * blockDim.x + threadIdx.x;   // E*8
    if (t >= Ee * 8) return;
    int e = t >> 3, q = t & 7;
    int s = src[e], d = dst[e];
    float w = dis[s] * dis[d];
    const float4 hv = *(const float4*)(h + (size_t)s * Hc + q * 4);
    float* ap = agg + (size_t)d * Hc + q * 4;
    atomicAdd(ap + 0, w * hv.x);
    atomicAdd(ap + 1, w * hv.y);
    atomicAdd(ap + 2, w * hv.z);
    atomicAdd(ap + 3, w * hv.w);
}

// ---------------- edge scatter (1-ch, layer 3) ----------------
__global__ void k_edge_scatter1(const float* __restrict__ h, const float* __restrict__ dis,
                                const int* __restrict__ src, const int* __restrict__ dst,
                                float* __restrict__ agg) {
    int e = blockIdx.x * blockDim.x + threadIdx.x;
    if (e >= Ee) return;
    int s = src[e], d = dst[e];
    atomicAdd(&agg[d], dis[s] * dis[d] * h[s]);
}

// ---------------- finalize: feat[:, off+c] = tanh(agg + dis^2*h + b) ----------------
template <int C>
__global__ void k_finalize(const float* __restrict__ agg, const float* __restrict__ h,
                           const float* __restrict__ dis, const float* __restrict__ bias,
                           float* __restrict__ feat, int off) {
    int t = blockIdx.x * blockDim.x + threadIdx.x;
    if (t >= Nn * C) return;
    int i = t / C, c = t - i * C;
    float di = dis[i];
    float v = agg[t] + di * di * h[t] + bias[c];
    feat[(size_t)i * FTc + off + c] = tanhf(v);
}

// ---------------- global_sort_pool: top-K by last channel, stable desc ----------------
__global__ void k_sortpool(const float* __restrict__ feat, float* __restrict__ topk) {
    __shared__ float vals[NPGc];
    int g = blockIdx.x;
    const float* fg = feat + (size_t)g * NPGc * FTc;
    for (int i = threadIdx.x; i < NPGc; i += blockDim.x)
        vals[i] = fg[(size_t)i * FTc + (FTc - 1)];
    __syncthreads();
    for (int i = threadIdx.x; i < NPGc; i += blockDim.x) {
        float v = vals[i];
        int rank = 0;
        for (int j = 0; j < NPGc; ++j) {
            float u = vals[j];
            rank += (u > v) || (u == v && j < i);   // argsort(-x) stability
        }
        if (rank < Kc) {
            float* dp = topk + ((size_t)g * Kc + rank) * FTc;
            const float* sp = fg + (size_t)i * FTc;
            for (int f = 0; f < FTc; ++f) dp[f] = sp[f];
        }
    }
}

// ---------------- conv1 (kernel=FT, stride=FT) + ReLU ----------------
__global__ void k_conv1(const float* __restrict__ topk, const float* __restrict__ w,
                        const float* __restrict__ b, float* __restrict__ out) {
    int t = blockIdx.x * blockDim.x + threadIdx.x;      // G*C1*P1
    if (t >= Gg * C1c * P1c) return;
    int p = t % P1c, oc = (t / P1c) % C1c, g = t / (P1c * C1c);
    const float* xr = topk + ((size_t)g * Kc + p) * FTc;
    const float* wr = w + oc * FTc;
    float s = b[oc];
    for (int f = 0; f < FTc; ++f) s += wr[f] * xr[f];
    out[t] = fmaxf(s, 0.f);
}

// ---------------- maxpool 1x2 stride 2 ----------------
__global__ void k_maxpool(const float* __restrict__ in, float* __restrict__ out) {
    int t = blockIdx.x * blockDim.x + threadIdx.x;      // G*C1*PP
    if (t >= Gg * C1c * PPc) return;
    int q = t % PPc, oc = (t / PPc) % C1c, g = t / (PPc * C1c);
    const float* ir = in + ((size_t)g * C1c + oc) * P1c;
    out[t] = fmaxf(ir[2 * q], ir[2 * q + 1]);
}

// ---------------- conv2 (C1->C2, kernel 5) + ReLU ----------------
__global__ void k_conv2(const float* __restrict__ in, const float* __restrict__ w,
                        const float* __restrict__ b, float* __restrict__ out) {
    int t = blockIdx.x * blockDim.x + threadIdx.x;      // G*C2*P2
    if (t >= Gg * C2c * P2c) return;
    int p = t % P2c, oc = (t / P2c) % C2c, g = t / (P2c * C2c);
    float s = b[oc];
    for (int ic = 0; ic < C1c; ++ic) {
        const float* ir = in + ((size_t)g * C1c + ic) * PPc + p;
        const float* wr = w + ((size_t)oc * C1c + ic) * KW2c;
#pragma unroll
        for (int kw = 0; kw < KW2c; ++kw) s += wr[kw] * ir[kw];
    }
    out[((size_t)g * C2c + oc) * P2c + p] = fmaxf(s, 0.f);
}

// ---------------- lin1 (352->128) + ReLU ----------------
__global__ void k_lin1(const float* __restrict__ x, const float* __restrict__ w,
                       const float* __restrict__ b, float* __restrict__ out) {
    int t = blockIdx.x * blockDim.x + threadIdx.x;      // G*128
    if (t >= Gg * 128) return;
    int j = t & 127, g = t >> 7;
    const float* xr = x + (size_t)g * DENSEc;
    float s = b[j];
    for (int d = 0; d < DENSEc; ++d) s += xr[d] * w[(size_t)d * 128 + j];
    out[t] = fmaxf(s, 0.f);
}

// ---------------- lin2 (128->1) ----------------
__global__ void k_lin2(const float* __restrict__ x, const float* __restrict__ w,
                       const float* __restrict__ b, float* __restrict__ out) {
    int g = blockIdx.x * blockDim.x + threadIdx.x;
    if (g >= Gg) return;
    const float* xr = x + (size_t)g * 128;
    float s = b[0];
#pragma unroll 8
    for (int j = 0; j < 128; ++j) s += xr[j] * w[j];
    out[g] = s;
}

extern "C" void kernel_launch(void* const* d_in, const int* in_sizes, int n_in,
                              void* d_out, int out_size, void* d_ws, size_t ws_size,
                              hipStream_t stream) {
    (void)in_sizes; (void)n_in; (void)out_size; (void)ws_size;
    const int*   z       = (const int*)d_in[0];
    const int*   ei      = (const int*)d_in[1];          // [2][E] row-major
    const int*   srcIdx  = ei;
    const int*   dstIdx  = ei + Ee;
    const float* z_emb   = (const float*)d_in[3];
    const float* Wl[3]   = { (const float*)d_in[4], (const float*)d_in[6], (const float*)d_in[8] };
    const float* bl[3]   = { (const float*)d_in[5], (const float*)d_in[7], (const float*)d_in[9] };
    const float* W3      = (const float*)d_in[10];
    const float* b3      = (const float*)d_in[11];
    const float* conv1_w = (const float*)d_in[12];
    const float* conv1_b = (const float*)d_in[13];
    const float* conv2_w = (const float*)d_in[14];
    const float* conv2_b = (const float*)d_in[15];
    const float* lin1_w  = (const float*)d_in[16];
    const float* lin1_b  = (const float*)d_in[17];
    const float* lin2_w  = (const float*)d_in[18];
    const float* lin2_b  = (const float*)d_in[19];
    float* out = (float*)d_out;

    // workspace carve-up (floats)
    float* ws   = (float*)d_ws;
    float* dis  = ws;                               // N      (deg, then rsqrt in place)
    float* x0   = dis + Nn;                         // N*32
    float* hbuf = x0 + (size_t)Nn * Hc;             // N*32
    float* agg  = hbuf + (size_t)Nn * Hc;           // N*32
    float* feat = agg + (size_t)Nn * Hc;            // N*97
    float* topk = feat + (size_t)Nn * FTc;          // G*K*97
    float* c1   = topk + (size_t)Gg * Kc * FTc;     // G*16*30
    float* pool = c1 + (size_t)Gg * C1c * P1c;      // G*16*15
    float* c2   = pool + (size_t)Gg * C1c * PPc;    // G*32*11
    float* l1   = c2 + (size_t)Gg * C2c * P2c;      // G*128

    const int TB = 256;

    // degree -> dis = deg^{-1/2}
    k_deg_init <<<Nn / TB, TB, 0, stream>>>(dis);
    k_deg_accum<<<Ee / TB, TB, 0, stream>>>(dstIdx, dis);
    k_deg_rsqrt<<<Nn / TB, TB, 0, stream>>>(dis);

    // embedding
    k_embed<<<(Nn * Hc) / TB, TB, 0, stream>>>(z, z_emb, x0);

    // GCN layers 0..2 (H->H, exact f32 WMMA) writing feat columns [32l, 32l+32)
    const float* Xin[3] = { x0, feat + 0, feat + 32 };
    const int    ldin[3] = { Hc, FTc, FTc };
    for (int l = 0; l < 3; ++l) {
        k_gcn_matmul_wmma<<<Nn / 64, 128, 0, stream>>>(Xin[l], ldin[l], Wl[l], hbuf);
        hipMemsetAsync(agg, 0, (size_t)Nn * Hc * sizeof(float), stream);
        k_edge_scatter32<<<(Ee * 8) / TB, TB, 0, stream>>>(hbuf, dis, srcIdx, dstIdx, agg);
        k_finalize<Hc><<<(Nn * Hc) / TB, TB, 0, stream>>>(agg, hbuf, dis, bl[l], feat, l * Hc);
    }

    // GCN layer 3 (H->1) writing feat column 96
    k_matvec32<<<Nn / TB, TB, 0, stream>>>(feat + 64, FTc, W3, hbuf);
    hipMemsetAsync(agg, 0, (size_t)Nn * sizeof(float), stream);
    k_edge_scatter1<<<Ee / TB, TB, 0, stream>>>(hbuf, dis, srcIdx, dstIdx, agg);
    k_finalize<1><<<Nn / TB, TB, 0, stream>>>(agg, hbuf, dis, b3, feat, FTc - 1);

    // sort-pool + CNN/MLP head
    k_sortpool<<<Gg, 256, 0, stream>>>(feat, topk);
    k_conv1 <<<(Gg * C1c * P1c + TB - 1) / TB, TB, 0, stream>>>(topk, conv1_w, conv1_b, c1);
    k_maxpool<<<(Gg * C1c * PPc + TB - 1) / TB, TB, 0, stream>>>(c1, pool);
    k_conv2 <<<(Gg * C2c * P2c + TB - 1) / TB, TB, 0, stream>>>(pool, conv2_w, conv2_b, c2);
    k_lin1  <<<(Gg * 128) / TB, TB, 0, stream>>>(c2, lin1_w, lin1_b, l1);
    k_lin2  <<<1, 128, 0, stream>>>(l1, lin2_w, lin2_b, out);
}